// LSTM_32203664785687
// MI455X (gfx1250) — compile-verified
//
#include <hip/hip_runtime.h>
#include <stddef.h>

#define E_DIM 300
#define G4    1200      // 4*E
#define BATCH 64
#define SEQ   64
#define DSEQ  63
#define ZHV   12016
#define REC_BLOCKS 75   // N-tiles of the recurrent GEMM (1200/16); one per block

typedef __attribute__((ext_vector_type(2))) float v2f;
typedef __attribute__((ext_vector_type(8))) float v8f;

__device__ __forceinline__ float sigmoidf_(float x) { return 1.0f / (1.0f + __expf(-x)); }

// ---------------------------------------------------------------------------
// Device-wide arrive/spin barrier (persistent-kernel sync).
// bar[0] = arrival count, bar[1] = generation.
// ---------------------------------------------------------------------------
__device__ __forceinline__ void grid_sync(unsigned* bar, unsigned nblocks)
{
    __threadfence();          // release this wave's global stores to device scope
    __syncthreads();
    if (threadIdx.x == 0) {
        unsigned g = __hip_atomic_load(bar + 1, __ATOMIC_RELAXED, __HIP_MEMORY_SCOPE_AGENT);
        unsigned prev = __hip_atomic_fetch_add(bar, 1u, __ATOMIC_ACQ_REL, __HIP_MEMORY_SCOPE_AGENT);
        if (prev == nblocks - 1) {
            __hip_atomic_store(bar, 0u, __ATOMIC_RELAXED, __HIP_MEMORY_SCOPE_AGENT);
            __hip_atomic_fetch_add(bar + 1, 1u, __ATOMIC_RELEASE, __HIP_MEMORY_SCOPE_AGENT);
        } else {
            while (__hip_atomic_load(bar + 1, __ATOMIC_ACQUIRE, __HIP_MEMORY_SCOPE_AGENT) == g) {
                __builtin_amdgcn_s_sleep(1);
            }
        }
    }
    __syncthreads();
}

// ---------------------------------------------------------------------------
// Register-blocked, software-pipelined WMMA GEMM:
//   D(M,N) [+]= A(M,K) @ W(N,K)^T [+ bias(N)]
// Each wave32 owns a 2x<NBV> block of 16x16 C tiles. Per k-step the 2+NBV
// float2 fragment loads for step k+4 are issued (independent hoisted base
// pointers -> clauseable) while the 2*NBV WMMAs for step k execute, so the
// matrix pipe waits on loads issued a full iteration earlier, not on the
// load it just issued. 192 B / WMMA of L2 traffic at NBV=4.
// ISA fragment layout (CDNA5 ISA):
//   A 16x4 : lane L -> row L%16, K pair 2*(L/16)       (float2)
//   B 4x16 : lane L -> col L%16, K pair 2*(L/16)       (float2, tile of W^T)
//   C 16x16: lane L, vgpr v -> row v + 8*(L/16), col L%16 (float8)
// ---------------------------------------------------------------------------
template <int NBV>
__device__ __forceinline__ void gemm_tileblock(const float* __restrict__ ap0, int lda,
                                               const float* __restrict__ wp0, int K,
                                               const float* __restrict__ bias,
                                               float* __restrict__ D, int ldd,
                                               int crow0, int ccol0, int accumulate)
{
    const v8f vzero = {};
    v8f acc[2][NBV];
#pragma unroll
    for (int i = 0; i < 2; ++i)
#pragma unroll
        for (int j = 0; j < NBV; ++j) acc[i][j] = vzero;

    if (accumulate) {
#pragma unroll
        for (int i = 0; i < 2; ++i)
#pragma unroll
            for (int j = 0; j < NBV; ++j) {
                const float* dp = D + (size_t)(crow0 + i * 16) * ldd + (ccol0 + j * 16);
#pragma unroll
                for (int v = 0; v < 8; ++v) acc[i][j][v] = dp[(size_t)v * ldd];
            }
    }

    // hoisted independent base pointers (no dependent v_add address chain)
    const float* ap1 = ap0 + (size_t)16 * lda;
    const float* wpj[NBV];
#pragma unroll
    for (int j = 0; j < NBV; ++j) wpj[j] = wp0 + (size_t)j * 16 * K;

    // software pipeline: prologue loads k=0
    v2f a0c = *(const v2f*)(ap0);
    v2f a1c = *(const v2f*)(ap1);
    v2f bc[NBV];
#pragma unroll
    for (int j = 0; j < NBV; ++j) bc[j] = *(const v2f*)(wpj[j]);

#pragma unroll 2
    for (int k0 = 0; k0 < K - 4; k0 += 4) {
        // issue next-iteration fragment loads first
        v2f a0n = *(const v2f*)(ap0 + k0 + 4);
        v2f a1n = *(const v2f*)(ap1 + k0 + 4);
        v2f bn[NBV];
#pragma unroll
        for (int j = 0; j < NBV; ++j) bn[j] = *(const v2f*)(wpj[j] + k0 + 4);
        // compute on current fragments (loaded one iteration ago)
#pragma unroll
        for (int j = 0; j < NBV; ++j) {
            acc[0][j] = __builtin_amdgcn_wmma_f32_16x16x4_f32(false, a0c, false, bc[j],
                                                              (short)0, acc[0][j], false, false);
            acc[1][j] = __builtin_amdgcn_wmma_f32_16x16x4_f32(false, a1c, false, bc[j],
                                                              (short)0, acc[1][j], false, false);
        }
        a0c = a0n; a1c = a1n;
#pragma unroll
        for (int j = 0; j < NBV; ++j) bc[j] = bn[j];
    }
    // epilogue: last k-step
#pragma unroll
    for (int j = 0; j < NBV; ++j) {
        acc[0][j] = __builtin_amdgcn_wmma_f32_16x16x4_f32(false, a0c, false, bc[j],
                                                          (short)0, acc[0][j], false, false);
        acc[1][j] = __builtin_amdgcn_wmma_f32_16x16x4_f32(false, a1c, false, bc[j],
                                                          (short)0, acc[1][j], false, false);
    }

#pragma unroll
    for (int i = 0; i < 2; ++i)
#pragma unroll
        for (int j = 0; j < NBV; ++j) {
            float* dp = D + (size_t)(crow0 + i * 16) * ldd + (ccol0 + j * 16);
            const float bv = bias ? bias[ccol0 + j * 16] : 0.0f;
#pragma unroll
            for (int v = 0; v < 8; ++v) dp[(size_t)v * ldd] = acc[i][j][v] + bv;
        }
}

__global__ __launch_bounds__(128)
void LSTM_wmma_gemm_atw(const float* __restrict__ A, int lda,
                        const float* __restrict__ W,
                        const float* __restrict__ bias,
                        float* __restrict__ D, int ldd,
                        int M, int N, int K, int accumulate)
{
    const int lane = threadIdx.x & 31;
    const int wave = (blockIdx.x * blockDim.x + threadIdx.x) >> 5;
    const int ntiles = N >> 4;
    const int mblk = (M >> 4) >> 1;               // M tiles always even here
    const int nblk = (ntiles + 3) >> 2;
    if (wave >= mblk * nblk) return;              // wave-uniform: EXEC stays all-1s
    const int bm = wave / nblk, bn = wave % nblk;
    const int tm0 = bm * 2, tn0 = bn * 4;
    const int rem = ntiles - tn0;
    const int nvalid = rem >= 4 ? 4 : rem;

    const int r = lane & 15;
    const int khalf = (lane >> 4) << 1;           // 0 or 2
    const float* ap0 = A + (size_t)(tm0 * 16 + r) * lda + khalf;
    const float* wp0 = W + (size_t)(tn0 * 16 + r) * K + khalf;
    const int crow0 = tm0 * 16 + ((lane >> 4) << 3);
    const int ccol0 = tn0 * 16 + r;

    switch (nvalid) {   // wave-uniform dispatch; tail blocks are rare (1 of 19 / 1 of 188)
    case 4: gemm_tileblock<4>(ap0, lda, wp0, K, bias, D, ldd, crow0, ccol0, accumulate); break;
    case 3: gemm_tileblock<3>(ap0, lda, wp0, K, bias, D, ldd, crow0, ccol0, accumulate); break;
    case 2: gemm_tileblock<2>(ap0, lda, wp0, K, bias, D, ldd, crow0, ccol0, accumulate); break;
    default: gemm_tileblock<1>(ap0, lda, wp0, K, bias, D, ldd, crow0, ccol0, accumulate); break;
    }
}

// ---------------------------------------------------------------------------
// Persistent recurrent layer: runs ALL timesteps of one LSTM layer in a
// single launch. Grid = 75 blocks x 128 threads = 300 waves = one wave per
// 16x16 tile of the M=64, N=1200 recurrent GEMM. Each block stages its
// 16-column Whh slab (16 x 300 f32 = 19.2 KB) into LDS once; per step the
// k-loop reads A(h) fragments from L2 and B fragments via ds_load_b64.
// GEMM phase and pointwise cell phase alternate under a device-wide barrier.
// ---------------------------------------------------------------------------
__global__ __launch_bounds__(128)
void LSTM_layer_rec(const float* __restrict__ Whh,
                    float* __restrict__ gates,          // (B*seq, 4E), holds x@Wih^T+b
                    float* __restrict__ h, float* __restrict__ c,
                    float* __restrict__ xout,           // (B, seq, E) layer output
                    int seq, unsigned* __restrict__ bar)
{
    __shared__ float wslab[16 * E_DIM];
    const int ntile = blockIdx.x;                 // 0..74
    const int lane  = threadIdx.x & 31;
    const int mtile = threadIdx.x >> 5;           // 0..3 (M = 64 = 4 tiles)

    {   // stage Whh rows [ntile*16, ntile*16+16) x K into LDS
        const float* wsrc = Whh + (size_t)ntile * 16 * E_DIM;
        for (int i = threadIdx.x; i < 16 * E_DIM; i += 128) wslab[i] = wsrc[i];
    }
    __syncthreads();

    const int r = lane & 15;
    const int khalf = (lane >> 4) << 1;
    const float* ap = h + (size_t)(mtile * 16 + r) * E_DIM + khalf;
    const float* wp = &wslab[r * E_DIM + khalf];  // 8B-aligned, conflict-free (44r mod 64 perm)
    const int crow = mtile * 16 + ((lane >> 4) << 3);
    const int ccol = ntile * 16 + r;
    const int ldd  = seq * G4;
    const int tid  = blockIdx.x * 128 + threadIdx.x;   // 0..9599

    for (int t = 0; t < seq; ++t) {
        // ---- GEMM phase: gates[b*seq+t, :] += h @ Whh^T (this wave's tile) ----
        float* dp = gates + (size_t)t * G4 + (size_t)crow * ldd + ccol;
        v8f acc;
#pragma unroll
        for (int v = 0; v < 8; ++v) acc[v] = dp[(size_t)v * ldd];
#pragma unroll 5
        for (int k0 = 0; k0 < E_DIM; k0 += 4) {
            v2f a = *(const v2f*)(ap + k0);
            v2f b = *(const v2f*)(wp + k0);
            acc = __builtin_amdgcn_wmma_f32_16x16x4_f32(false, a, false, b,
                                                        (short)0, acc, false, false);
        }
#pragma unroll
        for (int v = 0; v < 8; ++v) dp[(size_t)v * ldd] = acc[v];

        grid_sync(bar, REC_BLOCKS);               // all gate tiles visible

        // ---- cell phase: 19200 elements over 9600 threads ----
        for (int idx = tid; idx < BATCH * E_DIM; idx += REC_BLOCKS * 128) {
            const int b = idx / E_DIM, e = idx - b * E_DIM;
            const float* g = gates + (size_t)(b * seq + t) * G4;
            const float gi = sigmoidf_(g[e]);
            const float gf = sigmoidf_(g[E_DIM + e]);
            const float gg = tanhf(g[2 * E_DIM + e]);
            const float go = sigmoidf_(g[3 * E_DIM + e]);
            const float cn = gf * c[idx] + gi * gg;
            const float hn = go * tanhf(cn);
            c[idx] = cn;
            h[idx] = hn;
            xout[(size_t)(b * seq + t) * E_DIM + e] = hn;
        }
        grid_sync(bar, REC_BLOCKS);               // h visible before next step's GEMM
    }
}

// Embedding gather: one block per token
__global__ void LSTM_embed(const int* __restrict__ ids, int id_ld, int seq,
                           const float* __restrict__ table, float* __restrict__ out)
{
    const int tok = blockIdx.x;                   // b*seq + t
    const int b = tok / seq, t = tok - b * seq;
    const int id = ids[b * id_ld + t];
    const float* src = table + (size_t)id * E_DIM;
    float* dst = out + (size_t)tok * E_DIM;
    for (int i = threadIdx.x; i < E_DIM; i += blockDim.x) dst[i] = src[i];
}

__global__ void LSTM_fill0(float* __restrict__ p, int n)
{
    const int i = blockIdx.x * blockDim.x + threadIdx.x;
    if (i < n) p[i] = 0.0f;
}

// In-place log_softmax over rows of length N; one 256-thread block per row.
__global__ void LSTM_logsoftmax(float* __restrict__ d, int N)
{
    __shared__ float red[256];
    float* p = d + (size_t)blockIdx.x * N;
    float m = -1e30f;
    for (int i = threadIdx.x; i < N; i += 256) m = fmaxf(m, p[i]);
    red[threadIdx.x] = m; __syncthreads();
    for (int s = 128; s > 0; s >>= 1) {
        if (threadIdx.x < s) red[threadIdx.x] = fmaxf(red[threadIdx.x], red[threadIdx.x + s]);
        __syncthreads();
    }
    m = red[0]; __syncthreads();
    float sum = 0.0f;
    for (int i = threadIdx.x; i < N; i += 256) sum += __expf(p[i] - m);
    red[threadIdx.x] = sum; __syncthreads();
    for (int s = 128; s > 0; s >>= 1) {
        if (threadIdx.x < s) red[threadIdx.x] += red[threadIdx.x + s];
        __syncthreads();
    }
    const float lse = m + __logf(red[0]);
    for (int i = threadIdx.x; i < N; i += 256) p[i] -= lse;
}

// Deterministic masked-NLL reduction in a single block (no float atomics).
__global__ void LSTM_loss(const float* __restrict__ lp, const int* __restrict__ zh,
                          float* __restrict__ out)
{
    __shared__ float ss[256], sc[256];
    float s = 0.0f, cnt = 0.0f;
    for (int idx = threadIdx.x; idx < BATCH * DSEQ; idx += 256) {
        const int b = idx / DSEQ, t = idx - b * DSEQ;
        const int tgt = zh[b * SEQ + t + 1];       // zh_target
        if (tgt != 0) { s -= lp[(size_t)idx * ZHV + tgt]; cnt += 1.0f; }
    }
    ss[threadIdx.x] = s; sc[threadIdx.x] = cnt; __syncthreads();
    for (int st = 128; st > 0; st >>= 1) {
        if (threadIdx.x < st) { ss[threadIdx.x] += ss[threadIdx.x + st];
                                sc[threadIdx.x] += sc[threadIdx.x + st]; }
        __syncthreads();
    }
    if (threadIdx.x == 0) out[0] = ss[0] / sc[0];
}

extern "C" void kernel_launch(void* const* d_in, const int* in_sizes, int n_in,
                              void* d_out, int out_size, void* d_ws, size_t ws_size,
                              hipStream_t stream)
{
    const int*   en_input = (const int*)d_in[0];
    const int*   zh_input = (const int*)d_in[1];
    const float* en_embed = (const float*)d_in[2];
    const float* zh_embed = (const float*)d_in[3];
    const float* enc_Wih  = (const float*)d_in[4];
    const float* enc_Whh  = (const float*)d_in[5];
    const float* enc_b    = (const float*)d_in[6];
    const float* dec_Wih  = (const float*)d_in[7];
    const float* dec_Whh  = (const float*)d_in[8];
    const float* dec_b    = (const float*)d_in[9];
    const float* fc_W     = (const float*)d_in[10];
    const float* fc_b     = (const float*)d_in[11];
    float* out = (float*)d_out;
    float* ws  = (float*)d_ws;

    // workspace layout (floats); ~30 MB
    float* xb0   = ws;                            // (B,S,E): 64*64*300
    float* xb1   = xb0 + 1228800;                 // ping-pong layer buffer
    float* gates = xb1 + 1228800;                 // (B*S, 4E) = 4096*1200
    float* hbuf  = gates + 4915200;               // enc h per layer: 2*19200
    float* cbuf  = hbuf + 2 * 19200;
    float* dh    = cbuf + 2 * 19200;              // dec h
    float* dc    = dh + 2 * 19200;                // dec c
    unsigned* bar = (unsigned*)(dc + 2 * 19200);  // grid barrier: 2 words

    auto launch_gemm = [&](const float* A, int lda, const float* W, const float* bias,
                           float* D, int ldd, int M, int N, int K, int accum) {
        const int waves  = ((M / 16) / 2) * ((N / 16 + 3) / 4);
        const int blocks = (waves + 3) / 4;       // 4 waves per 128-thread block
        LSTM_wmma_gemm_atw<<<blocks, 128, 0, stream>>>(A, lda, W, bias, D, ldd, M, N, K, accum);
    };

    LSTM_fill0<<<1, 32, 0, stream>>>((float*)bar, 2);   // zero barrier words

    // ---------------- encoder ----------------
    LSTM_embed<<<BATCH * SEQ, 128, 0, stream>>>(en_input, SEQ, SEQ, en_embed, xb0);
    LSTM_fill0<<<(2 * 19200 + 255) / 256, 256, 0, stream>>>(hbuf, 2 * 19200);
    LSTM_fill0<<<(2 * 19200 + 255) / 256, 256, 0, stream>>>(cbuf, 2 * 19200);
    for (int l = 0; l < 2; ++l) {
        const float* X = (l == 0) ? xb0 : xb1;
        float* Xo      = (l == 0) ? xb1 : xb0;
        // hoisted input projection for all timesteps: (4096,1200) + bias
        launch_gemm(X, E_DIM, enc_Wih + (size_t)l * G4 * E_DIM, enc_b + l * G4,
                    gates, G4, BATCH * SEQ, G4, E_DIM, 0);
        // all 64 recurrent steps in ONE persistent launch
        LSTM_layer_rec<<<REC_BLOCKS, 128, 0, stream>>>(
            enc_Whh + (size_t)l * G4 * E_DIM, gates,
            hbuf + l * 19200, cbuf + l * 19200, Xo, SEQ, bar);
    }

    // ---------------- decoder ----------------
    LSTM_embed<<<BATCH * DSEQ, 128, 0, stream>>>(zh_input, SEQ, DSEQ, zh_embed, xb0);
    hipMemcpyAsync(dh, hbuf, 2 * 19200 * sizeof(float), hipMemcpyDeviceToDevice, stream);
    hipMemcpyAsync(dc, cbuf, 2 * 19200 * sizeof(float), hipMemcpyDeviceToDevice, stream);
    for (int l = 0; l < 2; ++l) {
        const float* X = (l == 0) ? xb0 : xb1;
        float* Xo      = (l == 0) ? xb1 : xb0;
        launch_gemm(X, E_DIM, dec_Wih + (size_t)l * G4 * E_DIM, dec_b + l * G4,
                    gates, G4, BATCH * DSEQ, G4, E_DIM, 0);
        LSTM_layer_rec<<<REC_BLOCKS, 128, 0, stream>>>(
            dec_Whh + (size_t)l * G4 * E_DIM, gates,
            dh + l * 19200, dc + l * 19200, Xo, DSEQ, bar);
    }

    // ------- FC directly into d_out, in-place log_softmax, loss -------
    launch_gemm(xb0 /* decoder layer-1 output */, E_DIM, fc_W, fc_b,
                out, ZHV, BATCH * DSEQ, ZHV, E_DIM, 0);
    LSTM_logsoftmax<<<BATCH * DSEQ, 256, 0, stream>>>(out, ZHV);
    LSTM_loss<<<1, 256, 0, stream>>>(out, zh_input, out + (out_size - 1));
}